// SparseLinear_58729382805835
// MI455X (gfx1250) — compile-verified
//
#include <hip/hip_runtime.h>

// CDNA5 / gfx1250 sparse output-projection (constrained-decoding logits).
//   logits[b, :] = -inf;  logits[b, valid_idxs[b,k]] = dot(x[b], W[valid_idxs[b,k]]) + bias[...]
// Bandwidth-bound (~80 MB @ 23.3 TB/s ~= 3.5 us floor); f32 WMMA keeps exact
// fp32 accumulate while putting the dot products on the matrix pipe.

typedef __attribute__((ext_vector_type(2))) float v2f;
typedef __attribute__((ext_vector_type(8))) float v8f;

#define BATCH 32
#define DDIM  1024
#define VOCAB 128000
#define KVAL  512
#define WAVES 8   // waves per block; each wave owns one 16-index tile

// ---------------------------------------------------------------------------
// Kernel 1: fill logits with -inf (vectorized 16B stores).
// ---------------------------------------------------------------------------
__global__ __launch_bounds__(256) void fill_neginf_kernel(float* __restrict__ out, int n4) {
  int i = blockIdx.x * blockDim.x + threadIdx.x;
  if (i < n4) {
    const float ninf = -__builtin_inff();
    float4 v = {ninf, ninf, ninf, ninf};
    ((float4*)out)[i] = v;
  }
}

// ---------------------------------------------------------------------------
// Kernel 2: gathered GEMV via V_WMMA_F32_16X16X4_F32 + scatter.
//
// Wave layout (wave32):
//   A (16x4 f32): lane L<16 holds row M=L, K=0..1; lane L>=16 holds M=L-16,
//                 K=2..3  (two VGPRs per lane)  -> gathered weight rows.
//   B (4x16 f32): every column n is the same x-chunk, so each half-wave loads
//                 its K pair {x[d0+2h], x[d0+2h+1]} from LDS.
//   C (16x16 f32, 8 VGPRs): all N columns equal; M=0..7 live in lanes 0-15,
//                 M=8..15 in lanes 16-31 -> lanes 0 and 16 scatter 8 each.
// ---------------------------------------------------------------------------
__global__ __launch_bounds__(256) void sparse_logits_wmma_kernel(
    const float* __restrict__ x, const float* __restrict__ weight,
    const float* __restrict__ bias, const int* __restrict__ valid_idxs,
    float* __restrict__ out) {
  __shared__ float xs[DDIM];

  const int b   = blockIdx.y;
  const int tid = threadIdx.x;

  // Stage x[b] (4 KB) into LDS: 256 threads x 16B.
  ((float4*)xs)[tid] = ((const float4*)(x + (size_t)b * DDIM))[tid];
  __syncthreads();

  const int wave = tid >> 5;
  const int lane = tid & 31;
  const int half = lane >> 4;   // 0: K=0..1, 1: K=2..3
  const int m    = lane & 15;
  const int k0   = (blockIdx.x * WAVES + wave) * 16;

  // Each lane streams its gathered row; lanes L and L+16 cover one row
  // contiguously (16B per depth-4 step -> same cache line across steps).
  const int    vidx_m = valid_idxs[b * KVAL + k0 + m];
  const float* wrow   = weight + (size_t)vidx_m * DDIM + half * 2;
  const float* xrow   = xs + half * 2;

  v8f acc = {};
  for (int d0 = 0; d0 < DDIM; d0 += 32) {  // 8 WMMAs per iteration
    v2f a[8], bv[8];
#pragma unroll
    for (int c = 0; c < 8; ++c) {
      a[c]  = *(const v2f*)(wrow + d0 + 4 * c);  // global_load_b64 (gather)
      bv[c] = *(const v2f*)(xrow + d0 + 4 * c);  // ds_load_b64 (broadcast x)
    }
#pragma unroll
    for (int c = 0; c < 8; ++c) {
      // 8 args: (neg_a, A, neg_b, B, c_mod, C, reuse_a, reuse_b)
      acc = __builtin_amdgcn_wmma_f32_16x16x4_f32(
          false, a[c], false, bv[c], (short)0, acc, false, false);
    }
  }

  // Scatter: one lane per half-wave writes its 8 rows (all N columns of C are
  // identical, so column 0's copy in this lane is the full dot product).
  if (m == 0) {
#pragma unroll
    for (int v = 0; v < 8; ++v) {
      const int mm = half * 8 + v;
      const int vi = valid_idxs[b * KVAL + k0 + mm];
      out[(size_t)b * VOCAB + vi] = acc[v] + bias[vi];
    }
  }
}

// ---------------------------------------------------------------------------
// Host-side launch (graph-capture safe: only kernel launches on `stream`).
// Inputs: 0:x [32,1024] f32, 1:weight [128000,1024] f32, 2:bias [128000] f32,
//         3:valid_idxs [32,512] int, 4:next_nodes (unused by reference output)
// Output: [32,128000] f32.
// ---------------------------------------------------------------------------
extern "C" void kernel_launch(void* const* d_in, const int* in_sizes, int n_in,
                              void* d_out, int out_size, void* d_ws, size_t ws_size,
                              hipStream_t stream) {
  (void)in_sizes; (void)n_in; (void)d_ws; (void)ws_size;

  const float* x          = (const float*)d_in[0];
  const float* weight     = (const float*)d_in[1];
  const float* bias       = (const float*)d_in[2];
  const int*   valid_idxs = (const int*)d_in[3];
  float*       out        = (float*)d_out;

  const int n4 = out_size / 4;  // B*V is divisible by 4
  fill_neginf_kernel<<<(n4 + 255) / 256, 256, 0, stream>>>(out, n4);

  dim3 grid(KVAL / (16 * WAVES), BATCH);  // (4, 32)
  sparse_logits_wmma_kernel<<<grid, 256, 0, stream>>>(x, weight, bias, valid_idxs, out);
}